// FactorAtt_ConvRelPosEnc_37142877176281
// MI455X (gfx1250) — compile-verified
//
#include <hip/hip_runtime.h>
#include <math.h>

typedef __attribute__((ext_vector_type(2))) float v2f;
typedef __attribute__((ext_vector_type(8))) float v8f;

#define DIMC   512
#define NHEADS 8
#define HD     64
#define SEQ    3136
#define BATCH  16
#define IMG_H  56
#define IMG_W  56
#define BH     (BATCH * NHEADS)      // 128
#define ATT_SCALE 0.125f             // 64^-0.5

__device__ __forceinline__ v2f ld2(const float* __restrict__ p) {
    const float2 t = *reinterpret_cast<const float2*>(p);
    v2f r; r.x = t.x; r.y = t.y; return r;
}

// ---------------------------------------------------------------------------
// Kernel 1: qkv GEMM  C[M=B*N,1536] = x[M,512] @ w_qkv^T, scatter to q/k/v
// layouts [B,h,N,d].  One wave = 16 rows x 64 cols (4 wmma accumulators).
// ---------------------------------------------------------------------------
__global__ void qkv_gemm(const float* __restrict__ x, const float* __restrict__ wqkv,
                         float* __restrict__ qb, float* __restrict__ kb,
                         float* __restrict__ vb) {
    const int lane = threadIdx.x & 31;
    const int hi   = lane >> 4;       // 0/1: which lane half
    const int l16  = lane & 15;
    const int wid  = blockIdx.x * (blockDim.x >> 5) + (threadIdx.x >> 5);
    const int mtile = wid / 24;       // 3136 row tiles
    const int cg    = wid % 24;       // 24 col groups of 64
    const long m0 = (long)mtile * 16;

    const float* arow = x + (m0 + l16) * DIMC + 2 * hi;
    const float* brow[4];
#pragma unroll
    for (int t = 0; t < 4; ++t)
        brow[t] = wqkv + (long)(cg * 64 + 16 * t + l16) * DIMC + 2 * hi;

    v8f acc[4] = {};
    for (int k0 = 0; k0 < DIMC; k0 += 4) {
        if (k0 + 64 < DIMC) __builtin_prefetch(arow + k0 + 64, 0, 1);
        v2f a = ld2(arow + k0);
#pragma unroll
        for (int t = 0; t < 4; ++t) {
            v2f b = ld2(brow[t] + k0);
            acc[t] = __builtin_amdgcn_wmma_f32_16x16x4_f32(
                false, a, false, b, (short)0, acc[t], false, false);
        }
    }

    const int which = cg >> 3;        // 0=q,1=k,2=v
    const int h     = cg & 7;
    float* dstbuf = (which == 0) ? qb : (which == 1) ? kb : vb;
#pragma unroll
    for (int j = 0; j < 8; ++j) {
        const long m = m0 + j + 8 * hi;          // C row = vgpr + 8*half
        const int bb = (int)(m / SEQ);
        const int n  = (int)(m % SEQ);
        const long base = (((long)(bb * NHEADS + h)) * SEQ + n) * HD;
#pragma unroll
        for (int t = 0; t < 4; ++t)
            dstbuf[base + 16 * t + l16] = acc[t][j];
    }
}

// ---------------------------------------------------------------------------
// Kernel 2: per-(b,h,channel) softmax stats over the sequence dim.
// One wave32 per reduction; shuffle tree reduce.
// ---------------------------------------------------------------------------
__global__ void k_stats(const float* __restrict__ kb,
                        float* __restrict__ mx, float* __restrict__ sm) {
    const int lane = threadIdx.x & 31;
    const int wid  = blockIdx.x * (blockDim.x >> 5) + (threadIdx.x >> 5); // 0..8191
    const int bh = wid >> 6;
    const int c  = wid & 63;
    const float* base = kb + (long)bh * SEQ * HD + c;

    float m = -INFINITY;
    for (int n = lane; n < SEQ; n += 32) m = fmaxf(m, base[(long)n * HD]);
#pragma unroll
    for (int off = 16; off > 0; off >>= 1) m = fmaxf(m, __shfl_xor(m, off, 32));

    float s = 0.f;
    for (int n = lane; n < SEQ; n += 32) s += __expf(base[(long)n * HD] - m);
#pragma unroll
    for (int off = 16; off > 0; off >>= 1) s += __shfl_xor(s, off, 32);

    if (lane == 0) { mx[wid] = m; sm[wid] = s; }
}

// ---------------------------------------------------------------------------
// Kernel 3: kv[b,h] = softmax(k)^T @ v  (64x64 per bh).
// A fragment = exp(k - max) computed on the fly; rows scaled by 1/sum.
// Block = 4 waves; wave w owns rows 16w..16w+15, all 64 cols.
// ---------------------------------------------------------------------------
__global__ void kv_gemm(const float* __restrict__ kb, const float* __restrict__ vb,
                        const float* __restrict__ mx, const float* __restrict__ sm,
                        float* __restrict__ kvb) {
    const int lane = threadIdx.x & 31;
    const int hi = lane >> 4, l16 = lane & 15;
    const int w  = threadIdx.x >> 5;      // 0..3
    const int bh = blockIdx.x;            // 0..127
    const int c0 = 16 * w;

    const float mxl = mx[bh * HD + c0 + l16];
    const float* kbase = kb + (long)bh * SEQ * HD + (c0 + l16);
    const float* vbase = vb + (long)bh * SEQ * HD + l16;

    v8f acc[4] = {};
    for (int k0 = 0; k0 < SEQ; k0 += 4) {
        const int r0 = k0 + 2 * hi;
        v2f a;
        a.x = __expf(kbase[(long)r0 * HD] - mxl);
        a.y = __expf(kbase[(long)(r0 + 1) * HD] - mxl);
#pragma unroll
        for (int t = 0; t < 4; ++t) {
            v2f b;
            b.x = vbase[(long)r0 * HD + 16 * t];
            b.y = vbase[(long)(r0 + 1) * HD + 16 * t];
            acc[t] = __builtin_amdgcn_wmma_f32_16x16x4_f32(
                false, a, false, b, (short)0, acc[t], false, false);
        }
    }
#pragma unroll
    for (int j = 0; j < 8; ++j) {
        const int c = c0 + j + 8 * hi;
        const float inv = 1.0f / sm[bh * HD + c];
#pragma unroll
        for (int t = 0; t < 4; ++t)
            kvb[((long)bh * HD + c) * HD + 16 * t + l16] = acc[t][j] * inv;
    }
}

// ---------------------------------------------------------------------------
// Kernel 4: out_heads = SCALE * (q @ kv) + q * depthwise_conv(v) fused.
// Wave = 16 seq positions x 64 channels of one head. Writes [B,N,512] tmp.
// ---------------------------------------------------------------------------
__global__ void attn_combine(const float* __restrict__ qb, const float* __restrict__ vb,
                             const float* __restrict__ kvb,
                             const float* __restrict__ w3, const float* __restrict__ b3,
                             const float* __restrict__ w5, const float* __restrict__ b5,
                             const float* __restrict__ w7, const float* __restrict__ b7,
                             float* __restrict__ tmp) {
    const int lane = threadIdx.x & 31;
    const int hi = lane >> 4, l16 = lane & 15;
    const int w  = threadIdx.x >> 5;
    const int bh = blockIdx.y;
    const int b  = bh >> 3, h = bh & 7;
    const int n0 = (blockIdx.x * 4 + w) * 16;    // 196 strips per bh

    const float* qrow = qb + ((long)bh * SEQ + n0 + l16) * HD + 2 * hi;
    const float* kv0  = kvb + (long)bh * HD * HD + l16;

    v8f acc[4] = {};
    for (int k0 = 0; k0 < HD; k0 += 4) {
        v2f a = ld2(qrow + k0);
        const int r0 = k0 + 2 * hi;
#pragma unroll
        for (int t = 0; t < 4; ++t) {
            v2f bf;
            bf.x = kv0[(long)r0 * HD + 16 * t];
            bf.y = kv0[(long)(r0 + 1) * HD + 16 * t];
            acc[t] = __builtin_amdgcn_wmma_f32_16x16x4_f32(
                false, a, false, bf, (short)0, acc[t], false, false);
        }
    }

    // per-head depthwise conv params (window dict {3:2, 5:3, 7:3})
    const float* cw; const float* cb; int ks;
    if (h < 2)      { cw = w3 + (long)(h)     * HD * 9;  cb = b3 + (h)     * HD; ks = 3; }
    else if (h < 5) { cw = w5 + (long)(h - 2) * HD * 25; cb = b5 + (h - 2) * HD; ks = 5; }
    else            { cw = w7 + (long)(h - 5) * HD * 49; cb = b7 + (h - 5) * HD; ks = 7; }
    const int pad = ks >> 1;
    const float* vbase = vb + (long)bh * SEQ * HD;

#pragma unroll
    for (int t = 0; t < 4; ++t) {
        const int c = 16 * t + l16;
        const float* wc = cw + c * ks * ks;
        const float bias = cb[c];
#pragma unroll
        for (int j = 0; j < 8; ++j) {
            const int n = n0 + j + 8 * hi;
            const int y = n / IMG_W, xp = n % IMG_W;
            float conv = bias;
            for (int dy = 0; dy < ks; ++dy) {
                const int yy = y + dy - pad;
                if (yy < 0 || yy >= IMG_H) continue;
                for (int dx = 0; dx < ks; ++dx) {
                    const int xx = xp + dx - pad;
                    if (xx < 0 || xx >= IMG_W) continue;
                    conv = fmaf(vbase[((long)yy * IMG_W + xx) * HD + c],
                                wc[dy * ks + dx], conv);
                }
            }
            const float qv = qb[((long)bh * SEQ + n) * HD + c];
            tmp[((long)b * SEQ + n) * DIMC + h * HD + c] =
                ATT_SCALE * acc[t][j] + qv * conv;
        }
    }
}

// ---------------------------------------------------------------------------
// Kernel 5: final projection  out[M,512] = tmp[M,512] @ w_proj^T + b_proj
// ---------------------------------------------------------------------------
__global__ void proj_gemm(const float* __restrict__ tmp, const float* __restrict__ wproj,
                          const float* __restrict__ bproj, float* __restrict__ out) {
    const int lane = threadIdx.x & 31;
    const int hi = lane >> 4, l16 = lane & 15;
    const int wid = blockIdx.x * (blockDim.x >> 5) + (threadIdx.x >> 5);
    const int mtile = wid >> 3;      // 3136 row tiles
    const int cg    = wid & 7;       // 8 col groups of 64
    const long m0 = (long)mtile * 16;

    const float* arow = tmp + (m0 + l16) * DIMC + 2 * hi;
    const float* brow[4];
#pragma unroll
    for (int t = 0; t < 4; ++t)
        brow[t] = wproj + (long)(cg * 64 + 16 * t + l16) * DIMC + 2 * hi;

    v8f acc[4] = {};
    for (int k0 = 0; k0 < DIMC; k0 += 4) {
        if (k0 + 64 < DIMC) __builtin_prefetch(arow + k0 + 64, 0, 1);
        v2f a = ld2(arow + k0);
#pragma unroll
        for (int t = 0; t < 4; ++t) {
            v2f b = ld2(brow[t] + k0);
            acc[t] = __builtin_amdgcn_wmma_f32_16x16x4_f32(
                false, a, false, b, (short)0, acc[t], false, false);
        }
    }
#pragma unroll
    for (int t = 0; t < 4; ++t) {
        const int gc = cg * 64 + 16 * t + l16;
        const float bias = bproj[gc];
#pragma unroll
        for (int j = 0; j < 8; ++j) {
            const long m = m0 + j + 8 * hi;
            out[m * DIMC + gc] = acc[t][j] + bias;
        }
    }
}

// ---------------------------------------------------------------------------
extern "C" void kernel_launch(void* const* d_in, const int* in_sizes, int n_in,
                              void* d_out, int out_size, void* d_ws, size_t ws_size,
                              hipStream_t stream) {
    (void)in_sizes; (void)n_in; (void)out_size; (void)ws_size;
    const float* x      = (const float*)d_in[0];
    // d_in[1]=H(56), d_in[2]=W(56) — constants baked in
    const float* w_qkv  = (const float*)d_in[3];
    const float* w_proj = (const float*)d_in[4];
    const float* b_proj = (const float*)d_in[5];
    const float* w3     = (const float*)d_in[6];
    const float* b3     = (const float*)d_in[7];
    const float* w5     = (const float*)d_in[8];
    const float* b5     = (const float*)d_in[9];
    const float* w7     = (const float*)d_in[10];
    const float* b7     = (const float*)d_in[11];
    float* out = (float*)d_out;

    float* ws = (float*)d_ws;
    const long QS = (long)BH * SEQ * HD;     // 25,690,112 floats per buffer
    float* qb  = ws;
    float* kb  = ws + QS;
    float* vb  = ws + 2 * QS;
    float* kvb = ws + 3 * QS;                // 128*64*64
    float* mx  = kvb + (long)BH * HD * HD;   // 8192
    float* sm  = mx + BH * HD;               // 8192
    float* tmp = kb;                         // kb is dead after kv_gemm: reuse

    // 1) qkv GEMM: 3136 mtiles * 24 col groups = 75264 waves, 8 waves/block
    qkv_gemm<<<9408, 256, 0, stream>>>(x, w_qkv, qb, kb, vb);
    // 2) softmax stats: 8192 waves
    k_stats<<<1024, 256, 0, stream>>>(kb, mx, sm);
    // 3) kv GEMM: one block of 4 waves per (b,h)
    kv_gemm<<<BH, 128, 0, stream>>>(kb, vb, mx, sm, kvb);
    // 4) factor_att + fused conv rel-pos: 196 strips per bh, 4 waves/block
    attn_combine<<<dim3(49, BH), 128, 0, stream>>>(qb, vb, kvb,
                                                   w3, b3, w5, b5, w7, b7, tmp);
    // 5) projection: 3136 mtiles * 8 col groups = 25088 waves, 8/block
    proj_gemm<<<3136, 256, 0, stream>>>(tmp, w_proj, b_proj, out);
}